// DeformableCrossAttention_42520176230418
// MI455X (gfx1250) — compile-verified
//
#include <hip/hip_runtime.h>
#include <hip/hip_bf16.h>
#include <math.h>

typedef float v2f __attribute__((ext_vector_type(2)));
typedef float v8f __attribute__((ext_vector_type(8)));

#define HW    9216      // 96*96
#define CCH   128
#define WIMG  96
#define NPIX  73728     // B*HW

#if __has_builtin(__builtin_amdgcn_global_load_async_to_lds_b128) && \
    __has_builtin(__builtin_amdgcn_s_wait_asynccnt)
#define USE_ASYNC_LDS 1
typedef int v4i_t __attribute__((vector_size(4 * sizeof(int))));
typedef __attribute__((address_space(1))) v4i_t* gp4i_t;   // global int4*
typedef __attribute__((address_space(3))) v4i_t* lp4i_t;   // LDS int4*
#endif

// -----------------------------------------------------------------------------
// Generic per-pixel 128x128 projection: out[o,p] = sum_k W[o*128+k]*in[k,p] (+bias)
// Block = 256 thr = 8 waves; wave w owns out-channel tile [16w,16w+16).
// Each block handles 8 consecutive 16-pixel tiles; B tile staged in LDS
// (async global->LDS DMA path when the toolchain exposes it).
// -----------------------------------------------------------------------------
__global__ __launch_bounds__(256) void gemm128_kernel(const float* __restrict__ in_,
                                                      const float* __restrict__ Wt,
                                                      const float* __restrict__ bias,
                                                      float* __restrict__ out_)
{
    __shared__ __align__(16) float ldsB[128 * 16];
    const int lane = threadIdx.x & 31;
    const int wv   = threadIdx.x >> 5;      // M tile 0..7
    const int n    = lane & 15;             // matrix column (and A row) index
    const int kh   = lane >> 4;             // K half-select
    const int mrow = 16 * wv + n;           // A-matrix row for this lane

    // per-thread staging slice: 8 consecutive LDS elements, all in one K-row
    const int e0 = threadIdx.x * 8;         // e0 % 16 is 0 or 8 -> single K row
    const int ek = e0 >> 4;                 // K row of this slice
    const int en = e0 & 15;                 // N offset of this slice

    for (int nt = 0; nt < 8; ++nt) {
        const int pixTile = blockIdx.x * 128 + nt * 16;
        const int b   = pixTile / HW;
        const int pix = pixTile - b * HW;
        const size_t inBase = (size_t)b * CCH * HW + (size_t)pix;
        const float* gsrc = in_ + inBase + (size_t)ek * HW + en;

#ifdef USE_ASYNC_LDS
        // async DMA of the 128(K) x 16(N) tile straight into LDS (2 x B128 / thread)
        {
            gp4i_t g0 = (gp4i_t)(unsigned long long)(uintptr_t)gsrc;
            gp4i_t g1 = (gp4i_t)(unsigned long long)(uintptr_t)(gsrc + 4);
            lp4i_t l0 = (lp4i_t)(unsigned int)(uintptr_t)&ldsB[e0];
            lp4i_t l1 = (lp4i_t)(unsigned int)(uintptr_t)&ldsB[e0 + 4];
            __builtin_amdgcn_global_load_async_to_lds_b128(g0, l0, 0, 0);
            __builtin_amdgcn_global_load_async_to_lds_b128(g1, l1, 0, 0);
            __builtin_amdgcn_s_wait_asynccnt(0);
        }
#else
        #pragma unroll
        for (int i = 0; i < 8; ++i)
            ldsB[e0 + i] = gsrc[i];
#endif
        // prefetch next tile's slice while this tile is being consumed
        if (nt < 7)
            __builtin_prefetch(gsrc + 16, 0, 0);
        __syncthreads();

        v8f acc = {};
        #pragma unroll
        for (int kk = 0; kk < 32; ++kk) {
            const int k0 = 4 * kk + 2 * kh;
            v2f a, bf;
            a.x  = Wt[mrow * 128 + k0];
            a.y  = Wt[mrow * 128 + k0 + 1];
            bf.x = ldsB[(k0    ) * 16 + n];
            bf.y = ldsB[(k0 + 1) * 16 + n];
            acc = __builtin_amdgcn_wmma_f32_16x16x4_f32(false, a, false, bf,
                                                        (short)0, acc, false, false);
        }

        #pragma unroll
        for (int r = 0; r < 8; ++r) {
            const int oc = 16 * wv + r + kh * 8;
            float vo = acc[r];
            if (bias) vo += bias[oc];
            out_[(size_t)b * CCH * HW + (size_t)oc * HW + pix + n] = vo;
        }
        __syncthreads();
    }
}

// -----------------------------------------------------------------------------
// 3x3 conv 128 -> 64 ('SAME', zero border) + bias + ReLU, as 9 shifted GEMM taps.
// Block = 8 waves: wave = (mt 0..3) x (pixel-subtile 0..1); 32 pixels / block.
// -----------------------------------------------------------------------------
__global__ __launch_bounds__(256) void conv3_kernel(const float* __restrict__ q,
                                                    const float* __restrict__ Wo1,
                                                    const float* __restrict__ bo1,
                                                    float* __restrict__ h1)
{
    const int lane = threadIdx.x & 31;
    const int wv   = threadIdx.x >> 5;
    const int mt   = wv & 3;                // out-channel tile (64 = 4 x 16)
    const int nt   = wv >> 2;               // pixel sub-tile
    const int n    = lane & 15;
    const int kh   = lane >> 4;

    const int pixTile = blockIdx.x * 32 + nt * 16;
    const int b   = pixTile / HW;
    const int rem = pixTile - b * HW;
    const int py  = rem / WIMG;
    const int px0 = rem - py * WIMG;
    const int mrow = 16 * mt + n;

    v8f acc = {};
    for (int ty = 0; ty < 3; ++ty) {
        const int yy  = py + ty - 1;
        const bool yok = (yy >= 0) && (yy < WIMG);
        for (int tx = 0; tx < 3; ++tx) {
            const int xx  = px0 + n + tx - 1;
            const bool ok = yok && (xx >= 0) && (xx < WIMG);
            const size_t qbase = (size_t)b * CCH * HW + (size_t)(yy * WIMG + xx);
            #pragma unroll
            for (int kk = 0; kk < 32; ++kk) {
                const int k0 = 4 * kk + 2 * kh;
                v2f a, bf;
                a.x  = Wo1[((mrow * 128 + k0    ) * 3 + ty) * 3 + tx];
                a.y  = Wo1[((mrow * 128 + k0 + 1) * 3 + ty) * 3 + tx];
                bf.x = ok ? q[qbase + (size_t)(k0    ) * HW] : 0.0f;
                bf.y = ok ? q[qbase + (size_t)(k0 + 1) * HW] : 0.0f;
                acc = __builtin_amdgcn_wmma_f32_16x16x4_f32(false, a, false, bf,
                                                            (short)0, acc, false, false);
            }
        }
    }
    #pragma unroll
    for (int r = 0; r < 8; ++r) {
        const int oc = 16 * mt + r + kh * 8;
        float vv = acc[r] + bo1[oc];
        vv = vv > 0.0f ? vv : 0.0f;
        h1[(size_t)b * 64 * HW + (size_t)oc * HW + rem + n] = vv;
    }
}

// -----------------------------------------------------------------------------
// Offset head (only the first x/y pair of Wo2 matters) + bilinear border sample.
// One wave per pixel; 64-wide dot reduced across lanes via shfl_xor.
// -----------------------------------------------------------------------------
__global__ __launch_bounds__(256) void offset_sample_kernel(const float* __restrict__ h1,
                                                            const float* __restrict__ Wo2,
                                                            const float* __restrict__ bo2,
                                                            const float* __restrict__ kv,
                                                            float* __restrict__ kvs)
{
    const int lane = threadIdx.x & 31;
    const int wv   = threadIdx.x >> 5;
    const int p    = blockIdx.x * 8 + wv;
    const int b    = p / HW;
    const int rem  = p - b * HW;
    const int py   = rem / WIMG;
    const int px   = rem - py * WIMG;

    const size_t h1base = (size_t)b * 64 * HW + rem;
    const float h0  = h1[h1base + (size_t)(2 * lane    ) * HW];
    const float h1v = h1[h1base + (size_t)(2 * lane + 1) * HW];
    float ox = Wo2[     2 * lane] * h0 + Wo2[     2 * lane + 1] * h1v;
    float oy = Wo2[64 + 2 * lane] * h0 + Wo2[64 + 2 * lane + 1] * h1v;
    #pragma unroll
    for (int off = 16; off > 0; off >>= 1) {
        ox += __shfl_xor(ox, off, 32);
        oy += __shfl_xor(oy, off, 32);
    }
    ox = (ox + bo2[0]) * 0.1f;
    oy = (oy + bo2[1]) * 0.1f;

    // ix = clip((gx+1)*(W-1)/2); gx = (2px/(W-1)-1) + ox  ==>  ix = px + ox*(W-1)/2
    float ix = (float)px + ox * 47.5f;
    float iy = (float)py + oy * 47.5f;
    ix = fminf(fmaxf(ix, 0.0f), 95.0f);
    iy = fminf(fmaxf(iy, 0.0f), 95.0f);
    const float x0f = floorf(ix), y0f = floorf(iy);
    const float wx = ix - x0f, wy = iy - y0f;
    const int x0 = (int)x0f, y0 = (int)y0f;
    const int x1 = min(x0 + 1, 95), y1 = min(y0 + 1, 95);
    const float w00 = (1.0f - wx) * (1.0f - wy);
    const float w01 = wx * (1.0f - wy);
    const float w10 = (1.0f - wx) * wy;
    const float w11 = wx * wy;

    const size_t kvb = (size_t)b * CCH * HW;
    #pragma unroll
    for (int ci = 0; ci < 4; ++ci) {
        const int c = lane + 32 * ci;
        const size_t cb = kvb + (size_t)c * HW;
        const float v00 = kv[cb + y0 * WIMG + x0];
        const float v01 = kv[cb + y0 * WIMG + x1];
        const float v10 = kv[cb + y1 * WIMG + x0];
        const float v11 = kv[cb + y1 * WIMG + x1];
        kvs[cb + rem] = v00 * w00 + v01 * w01 + v10 * w10 + v11 * w11;
    }
}

// -----------------------------------------------------------------------------
// Per-pixel attention ACROSS heads: q,k,v are (8 heads x 16 dh) per pixel.
// One wave per pixel; lane = (head = lane>>2, d-quad = lane&3).
// -----------------------------------------------------------------------------
__global__ __launch_bounds__(256) void attn_kernel(const float* __restrict__ q,
                                                   const float* __restrict__ k,
                                                   const float* __restrict__ v,
                                                   float* __restrict__ o)
{
    const int lane = threadIdx.x & 31;
    const int wv   = threadIdx.x >> 5;
    const int p    = blockIdx.x * 8 + wv;
    const int b    = p / HW;
    const int rem  = p - b * HW;
    const size_t base = (size_t)b * CCH * HW + rem;

    float qv[4], kv_[4], vv[4];
    #pragma unroll
    for (int i = 0; i < 4; ++i) {
        const size_t a = base + (size_t)(4 * lane + i) * HW;
        qv[i] = q[a];  kv_[i] = k[a];  vv[i] = v[a];
    }
    const float scale = 0.25f;   // dh=16 -> 16^-0.5

    float lg[8];
    #pragma unroll
    for (int g = 0; g < 8; ++g) {
        const int src = 4 * g + (lane & 3);
        float part = 0.0f;
        #pragma unroll
        for (int i = 0; i < 4; ++i)
            part += qv[i] * __shfl(kv_[i], src, 32);
        part += __shfl_xor(part, 1, 32);
        part += __shfl_xor(part, 2, 32);
        lg[g] = part * scale;
    }
    float mx = lg[0];
    #pragma unroll
    for (int g = 1; g < 8; ++g) mx = fmaxf(mx, lg[g]);
    float s = 0.0f;
    #pragma unroll
    for (int g = 0; g < 8; ++g) { lg[g] = __expf(lg[g] - mx); s += lg[g]; }
    const float inv = 1.0f / s;

    float acc[4] = {0.0f, 0.0f, 0.0f, 0.0f};
    #pragma unroll
    for (int g = 0; g < 8; ++g) {
        const int src = 4 * g + (lane & 3);
        const float a = lg[g] * inv;
        #pragma unroll
        for (int i = 0; i < 4; ++i)
            acc[i] += a * __shfl(vv[i], src, 32);
    }
    #pragma unroll
    for (int i = 0; i < 4; ++i)
        o[base + (size_t)(4 * lane + i) * HW] = acc[i];
}

// -----------------------------------------------------------------------------
extern "C" void kernel_launch(void* const* d_in, const int* in_sizes, int n_in,
                              void* d_out, int out_size, void* d_ws, size_t ws_size,
                              hipStream_t stream) {
    (void)in_sizes; (void)n_in; (void)out_size; (void)ws_size;

    const float* query_map = (const float*)d_in[0];
    const float* kv_map    = (const float*)d_in[1];
    const float* Wq        = (const float*)d_in[2];
    const float* Wo1       = (const float*)d_in[3];
    const float* bo1       = (const float*)d_in[4];
    const float* Wo2       = (const float*)d_in[5];
    const float* bo2       = (const float*)d_in[6];
    const float* Wk        = (const float*)d_in[7];
    const float* Wv        = (const float*)d_in[8];
    const float* Wout      = (const float*)d_in[9];
    const float* bout      = (const float*)d_in[10];
    float*       out       = (float*)d_out;

    // workspace layout (floats)
    float* ws   = (float*)d_ws;
    float* qb   = ws;                               // 128ch : 9437184
    float* h1b  = qb  + (size_t)NPIX * 128;         //  64ch : 4718592
    float* kvsb = h1b + (size_t)NPIX * 64;          // 128ch : 9437184 (reused as attn out)
    float* kb   = kvsb + (size_t)NPIX * 128;        // 128ch
    float* vb   = kb  + (size_t)NPIX * 128;         // 128ch
    float* attb = kvsb;                             // alias: kvs dead after k/v GEMMs

    // 1. q = Wq * x
    gemm128_kernel<<<NPIX / 128, 256, 0, stream>>>(query_map, Wq, nullptr, qb);
    // 2. h1 = relu(conv3x3(q) + bo1)
    conv3_kernel<<<NPIX / 32, 256, 0, stream>>>(qb, Wo1, bo1, h1b);
    // 3. offsets (first pair) + bilinear border sample of kv_map
    offset_sample_kernel<<<NPIX / 8, 256, 0, stream>>>(h1b, Wo2, bo2, kv_map, kvsb);
    // 4/5. k = Wk * kvs ; v = Wv * kvs
    gemm128_kernel<<<NPIX / 128, 256, 0, stream>>>(kvsb, Wk, nullptr, kb);
    gemm128_kernel<<<NPIX / 128, 256, 0, stream>>>(kvsb, Wv, nullptr, vb);
    // 6. per-pixel cross-head attention
    attn_kernel<<<NPIX / 8, 256, 0, stream>>>(qb, kb, vb, attb);
    // 7. out = Wout * attn + bout
    gemm128_kernel<<<NPIX / 128, 256, 0, stream>>>(attb, Wout, bout, out);
}